// SelfAttention_32942399160710
// MI455X (gfx1250) — compile-verified
//
#include <hip/hip_runtime.h>
#include <hip/hip_bf16.h>
#include <math.h>

// ---------------------------------------------------------------------------
// Problem constants (from reference: B=8, C=256, H=W=64)
// ---------------------------------------------------------------------------
#define BATCH 8
#define CCH   256        // channels
#define NTOK  4096       // H*W
#define MTOK  1024       // (H/2)*(W/2)
#define DQK   32         // C/8
#define DV    128        // C/2

typedef float v2f __attribute__((ext_vector_type(2)));
typedef float v8f __attribute__((ext_vector_type(8)));

// Exact-fp32 CDNA5 WMMA: D(16x16) = A(16x4) * B(4x16) + C
// 8-arg VOP3P form: (neg_a, A, neg_b, B, c_mod, C, reuse_a, reuse_b)
__device__ __forceinline__ v8f wmma_f32_4(v2f a, v2f b, v8f c) {
  return __builtin_amdgcn_wmma_f32_16x16x4_f32(false, a, false, b, (short)0, c,
                                               false, false);
}

// A(16x4) lane layout: row = lane&15, VGPR0/1 hold K = 2*(lane>>4) + {0,1}
// B(4x16) lane layout: col = lane&15, VGPR0/1 hold K = 2*(lane>>4) + {0,1}
// D(16x16): vgpr i -> M = i + 8*(lane>>4), N = lane&15

// ---------------------------------------------------------------------------
// Kernel 1: generic projection GEMM  D[b][M][NTOK] = W[M][CCH] * X[b][CCH][NTOK]
// one wave per 16x16 output tile; 8 waves per block
// ---------------------------------------------------------------------------
__global__ __launch_bounds__(256) void proj_gemm(const float* __restrict__ Wm,
                                                 const float* __restrict__ X,
                                                 float* __restrict__ D,
                                                 int Mdim) {
  const int lane  = threadIdx.x & 31;
  const int wid   = threadIdx.x >> 5;
  const int nt_n  = NTOK / 16;                  // 256
  const int tiles = (Mdim / 16) * nt_n;
  const int t     = blockIdx.x * 8 + wid;       // wave-uniform
  if (t >= tiles) return;                       // whole wave exits together
  const int mt = t / nt_n;
  const int nt = t % nt_n;
  const int b  = blockIdx.y;

  const float* Xb = X + (size_t)b * CCH * NTOK;
  float*       Db = D + (size_t)b * Mdim * NTOK;

  const int r     = lane & 15;
  const int khalf = (lane >> 4) * 2;            // 0 or 2

  v8f acc = {};
  const float* wrow = Wm + (size_t)(mt * 16 + r) * CCH;
#pragma unroll 4
  for (int k0 = 0; k0 < CCH; k0 += 4) {
    v2f a = *(const v2f*)(wrow + k0 + khalf);   // contiguous pair -> b64 load
    v2f bm;
    bm.x = Xb[(size_t)(k0 + khalf) * NTOK + nt * 16 + r];
    bm.y = Xb[(size_t)(k0 + khalf + 1) * NTOK + nt * 16 + r];
    acc = wmma_f32_4(a, bm, acc);
  }
#pragma unroll
  for (int i = 0; i < 8; ++i) {
    const int mrow = mt * 16 + i + 8 * (lane >> 4);
    Db[(size_t)mrow * NTOK + nt * 16 + r] = acc[i];
  }
}

// ---------------------------------------------------------------------------
// Kernel 2: 2x2/stride-2 maxpool over 64x64 spatial, NCHW
// in: [BATCH][Cch][64][64] -> out: [BATCH][Cch][32][32]
// ---------------------------------------------------------------------------
__global__ __launch_bounds__(256) void maxpool2(const float* __restrict__ in,
                                                float* __restrict__ out,
                                                int Cch) {
  const size_t idx   = (size_t)blockIdx.x * blockDim.x + threadIdx.x;
  const size_t total = (size_t)BATCH * Cch * 32 * 32;
  if (idx >= total) return;
  const int    xo = idx & 31;
  const int    yo = (idx >> 5) & 31;
  const size_t bc = idx >> 10;
  const float* p  = in + bc * 4096 + (size_t)(yo * 2) * 64 + xo * 2;
  out[idx] = fmaxf(fmaxf(p[0], p[1]), fmaxf(p[64], p[65]));
}

// ---------------------------------------------------------------------------
// Kernel 3: fused attention. One block = 16 query rows of one batch.
//   S(16x1024) = Q^T * Kp     (WMMA, 8 waves over 64 m-tiles, into LDS)
//   P = softmax_rows(S)       (fp32, in LDS)
//   O(128 x 16) += Vp * P^T   (WMMA, wave w owns dv-tile w, 256 K-steps)
// Q:  [b][DQK][NTOK]   Kp: [b][DQK][MTOK]   Vp: [b][DV][MTOK]
// O:  [b][DV][NTOK]
// ---------------------------------------------------------------------------
__global__ __launch_bounds__(256) void attn_fused(const float* __restrict__ Q,
                                                  const float* __restrict__ Kp,
                                                  const float* __restrict__ Vp,
                                                  float* __restrict__ O) {
  extern __shared__ float smem[];
  float* S   = smem;               // 16 * MTOK floats (64 KB)
  float* red = smem + 16 * MTOK;   // 16 * 16 reduction scratch

  const int b  = blockIdx.y;
  const int n0 = blockIdx.x * 16;

  const int lane  = threadIdx.x & 31;
  const int wid   = threadIdx.x >> 5;   // 0..7
  const int r     = lane & 15;
  const int hi    = lane >> 4;          // 0/1
  const int khalf = hi * 2;

  const float* Qb = Q  + (size_t)b * DQK * NTOK;
  const float* Kb = Kp + (size_t)b * DQK * MTOK;
  const float* Vb = Vp + (size_t)b * DV * MTOK;
  float*       Ob = O  + (size_t)b * DV * NTOK;

  // ---- phase 1: score strip S[16][MTOK] ----
  for (int mt = wid; mt < MTOK / 16; mt += 8) {   // wave-uniform loop
    v8f acc = {};
#pragma unroll
    for (int k0 = 0; k0 < DQK; k0 += 4) {
      v2f a, bm;
      a.x  = Qb[(size_t)(k0 + khalf) * NTOK + n0 + r];
      a.y  = Qb[(size_t)(k0 + khalf + 1) * NTOK + n0 + r];
      bm.x = Kb[(size_t)(k0 + khalf) * MTOK + mt * 16 + r];
      bm.y = Kb[(size_t)(k0 + khalf + 1) * MTOK + mt * 16 + r];
      acc = wmma_f32_4(a, bm, acc);
    }
#pragma unroll
    for (int i = 0; i < 8; ++i)
      S[(i + 8 * hi) * MTOK + mt * 16 + r] = acc[i];
  }
  __syncthreads();

  // ---- phase 2: row softmax (16 rows, 16 threads per row) ----
  {
    const int row = threadIdx.x >> 4;   // 0..15
    const int tt  = threadIdx.x & 15;   // 0..15
    float* Srow   = S + (size_t)row * MTOK;

    float mx = -INFINITY;
    for (int j = tt; j < MTOK; j += 16) mx = fmaxf(mx, Srow[j]);
    red[row * 16 + tt] = mx;
    __syncthreads();
    if (tt == 0) {
      float m2 = red[row * 16];
      for (int j = 1; j < 16; ++j) m2 = fmaxf(m2, red[row * 16 + j]);
      red[row * 16] = m2;
    }
    __syncthreads();
    mx = red[row * 16];
    float s = 0.f;
    for (int j = tt; j < MTOK; j += 16) {
      float e = __expf(Srow[j] - mx);
      Srow[j] = e;
      s += e;
    }
    __syncthreads();
    red[row * 16 + tt] = s;
    __syncthreads();
    if (tt == 0) {
      float s2 = 0.f;
      for (int j = 0; j < 16; ++j) s2 += red[row * 16 + j];
      red[row * 16] = 1.0f / s2;
    }
    __syncthreads();
    const float inv = red[row * 16];
    for (int j = tt; j < MTOK; j += 16) Srow[j] *= inv;
  }
  __syncthreads();

  // ---- phase 3: O^T tile. D[16n x 16v] = P(16 x MTOK) * Vp^T(MTOK x 16) ----
  {
    const int v0 = wid * 16;   // 8 waves x 16 = DV
    v8f acc = {};
    for (int m0 = 0; m0 < MTOK; m0 += 4) {
      // A from LDS: P[r][m0+khalf .. +1]  (contiguous -> ds_load_b64)
      v2f a = *(const v2f*)(S + (size_t)r * MTOK + m0 + khalf);
      // B: B[kk][col] = Vp[v0+col][m0+kk]  (contiguous pair -> b64 load)
      v2f bm = *(const v2f*)(Vb + (size_t)(v0 + r) * MTOK + m0 + khalf);
      acc = wmma_f32_4(a, bm, acc);
    }
#pragma unroll
    for (int i = 0; i < 8; ++i) {
      const int nrow = n0 + i + 8 * hi;     // D row -> n index
      Ob[(size_t)(v0 + r) * NTOK + nrow] = acc[i];   // D col (lane&15) -> v
    }
  }
}

// ---------------------------------------------------------------------------
// Kernel 4: out = gamma * Wo(256x128) * O(b,128,NTOK) + x   (fused epilogue)
// ---------------------------------------------------------------------------
__global__ __launch_bounds__(256) void out_proj(const float* __restrict__ Wo,
                                                const float* __restrict__ O,
                                                const float* __restrict__ X,
                                                const float* __restrict__ gammap,
                                                float* __restrict__ out) {
  const int lane  = threadIdx.x & 31;
  const int wid   = threadIdx.x >> 5;
  const int nt_n  = NTOK / 16;                   // 256
  const int t     = blockIdx.x * 8 + wid;        // wave-uniform, 4096 tiles
  const int mt    = t / nt_n;
  const int nt    = t % nt_n;
  const int b     = blockIdx.y;

  const float* Ob  = O   + (size_t)b * DV * NTOK;
  const float* Xb  = X   + (size_t)b * CCH * NTOK;
  float*       Db  = out + (size_t)b * CCH * NTOK;
  const float gamma = gammap[0];

  const int r     = lane & 15;
  const int hi    = lane >> 4;
  const int khalf = hi * 2;

  v8f acc = {};
  const float* wrow = Wo + (size_t)(mt * 16 + r) * DV;
#pragma unroll 4
  for (int k0 = 0; k0 < DV; k0 += 4) {
    v2f a = *(const v2f*)(wrow + k0 + khalf);    // Wo row, contiguous
    v2f bm;
    bm.x = Ob[(size_t)(k0 + khalf) * NTOK + nt * 16 + r];
    bm.y = Ob[(size_t)(k0 + khalf + 1) * NTOK + nt * 16 + r];
    acc = wmma_f32_4(a, bm, acc);
  }
#pragma unroll
  for (int i = 0; i < 8; ++i) {
    const int    c   = mt * 16 + i + 8 * hi;
    const size_t off = (size_t)c * NTOK + nt * 16 + r;
    Db[off] = gamma * acc[i] + Xb[off];
  }
}

// ---------------------------------------------------------------------------
// Host-side launch
// ---------------------------------------------------------------------------
extern "C" void kernel_launch(void* const* d_in, const int* in_sizes, int n_in,
                              void* d_out, int out_size, void* d_ws, size_t ws_size,
                              hipStream_t stream) {
  const float* x     = (const float*)d_in[0];   // [8,256,64,64]
  const float* Wq    = (const float*)d_in[1];   // [32,256]
  const float* Wk    = (const float*)d_in[2];   // [32,256]
  const float* Wv    = (const float*)d_in[3];   // [128,256]
  const float* Wo    = (const float*)d_in[4];   // [256,128]
  const float* gamma = (const float*)d_in[5];   // scalar
  float*       out   = (float*)d_out;

  // workspace layout (floats): Q | Kpool | Vpool | O (O doubles as pre-pool tmp)
  float*       ws  = (float*)d_ws;
  float*       wsQ = ws;                                      // 8*32*4096
  float*       wsK = wsQ + (size_t)BATCH * DQK * NTOK;        // 8*32*1024
  float*       wsV = wsK + (size_t)BATCH * DQK * MTOK;        // 8*128*1024
  float*       wsO = wsV + (size_t)BATCH * DV * MTOK;         // 8*128*4096
  (void)in_sizes; (void)n_in; (void)out_size; (void)ws_size;

  // 1) Q projection: M=32 -> 512 tiles -> 64 blocks of 8 waves
  proj_gemm<<<dim3(64, BATCH), 256, 0, stream>>>(Wq, x, wsQ, DQK);

  // 2) K projection (pre-pool into wsO tmp), then pool -> wsK
  proj_gemm<<<dim3(64, BATCH), 256, 0, stream>>>(Wk, x, wsO, DQK);
  {
    size_t total = (size_t)BATCH * DQK * 32 * 32;   // 262144
    maxpool2<<<dim3((unsigned)((total + 255) / 256)), 256, 0, stream>>>(wsO, wsK, DQK);
  }

  // 3) V projection (pre-pool into wsO tmp), then pool -> wsV
  proj_gemm<<<dim3(256, BATCH), 256, 0, stream>>>(Wv, x, wsO, DV);
  {
    size_t total = (size_t)BATCH * DV * 32 * 32;    // 1048576
    maxpool2<<<dim3((unsigned)((total + 255) / 256)), 256, 0, stream>>>(wsO, wsV, DV);
  }

  // 4) fused attention: 256 query-row blocks per batch; 64KB+1KB dynamic LDS
  {
    const size_t shmem = (size_t)(16 * MTOK + 16 * 16) * sizeof(float); // 66560 B
    attn_fused<<<dim3(NTOK / 16, BATCH), 256, shmem, stream>>>(wsQ, wsK, wsV, wsO);
  }

  // 5) output projection + gamma-scaled residual: 4096 tiles -> 512 blocks
  out_proj<<<dim3(512, BATCH), 256, 0, stream>>>(Wo, wsO, x, gamma, out);
}